// YoloLayer_1288490188800
// MI455X (gfx1250) — compile-verified
//
#include <hip/hip_runtime.h>
#include <stdint.h>

// ---------------------------------------------------------------------------
// YOLO loss on MI455X (gfx1250, wave32).
// Memory-bound streaming reduction: ~200 MB in, 128 B out @ 23.3 TB/s ~ 9 us.
// Design: one wave32 per (b,h,w,a) cell, lanes stripe the 85 channels with
// coalesced b32 loads (340 B cell stride forbids wider aligned vectors);
// per-batch true_boxes (800 B) and anchors (72 B) are broadcast into LDS via
// the Tensor Data Mover (tensor_load_to_lds + s_wait_tensorcnt); butterfly
// __shfl_xor reductions for class-max / sum-exp / argmax / best-IoU.
// No matmul structure exists here, so WMMA is deliberately not used.
// ---------------------------------------------------------------------------

#define NB 32
#define NH 32
#define NW 32
#define NA 9
#define NC 80
#define NT 50
#define NCH 85                    // 5 + NC
#define CELLS_PER_ROW (NW * NA)   // 288 cells per (b,h) block
#define IGNORE_THRESH 0.5f
#define OBJ_SCALE 5.0f

typedef unsigned int u32;
typedef unsigned long long u64;
typedef u32 tdm_v4u __attribute__((ext_vector_type(4)));
typedef int  tdm_v4i __attribute__((ext_vector_type(4)));
typedef int  tdm_v8i __attribute__((ext_vector_type(8)));

// Build a 1-row TDM descriptor: copy `nelem` fp32 from `gptr` to LDS `ldsaddr`.
__device__ __forceinline__ void tdm_copy_row_f32(const float* gptr, u32 ldsaddr,
                                                 u32 nelem) {
  u64 gaddr = (u64)(const void*)gptr;
  tdm_v4u g0;
  g0[0] = 1u;                                   // count=1 (valid user descriptor)
  g0[1] = ldsaddr;                              // lds_addr  -> bits [63:32]
  g0[2] = (u32)(gaddr & 0xFFFFFFFFu);           // global_addr[31:0]
  g0[3] = (u32)((gaddr >> 32) & 0x01FFFFFFu)    // global_addr[56:32]
        | (2u << 30);                           // type=2 ("image")
  tdm_v8i g1;
  g1[0] = (int)(2u << 16);          // workgroup_mask=0 | data_size=2 (4 B)
  g1[1] = (int)(nelem << 16);       // tensor_dim0[15:0] at bits [63:48]
  g1[2] = (int)(1u << 16);          // tensor_dim0 hi=0 | tensor_dim1=1
  g1[3] = (int)(nelem << 16);       // tensor_dim1 hi=0 | tile_dim0=nelem
  g1[4] = 1;                        // tile_dim1=1, tile_dim2=0 (unused)
  g1[5] = (int)nelem;               // tensor_dim0_stride lo32
  g1[6] = (int)(nelem << 16);       // stride0 hi=0 | tensor_dim1_stride lo16
  g1[7] = 0;
  tdm_v4i gz = {0, 0, 0, 0};
#if defined(__clang_major__) && (__clang_major__ >= 23)
  tdm_v8i gz8 = {0, 0, 0, 0, 0, 0, 0, 0};
  __builtin_amdgcn_tensor_load_to_lds(g0, g1, gz, gz, gz8, 0);
#else
  __builtin_amdgcn_tensor_load_to_lds(g0, g1, gz, gz, 0);
#endif
}

__global__ void zero_out_kernel(float* __restrict__ out, int n) {
  int i = blockIdx.x * blockDim.x + threadIdx.x;
  if (i < n) out[i] = 0.0f;
}

__global__ __launch_bounds__(256) void yolo_loss_kernel(
    const float* __restrict__ y_pred,     // (B,H,W,A*85) fp32
    const float* __restrict__ y_true,     // (B,H,W,A,85) fp32 -- same flat layout
    const float* __restrict__ true_boxes, // (B,50,4) fp32
    const float* __restrict__ anchors,    // (18,) fp32
    float* __restrict__ out)              // (B,) fp32, pre-zeroed
{
  const int tid  = (int)threadIdx.x;
  const int lane = tid & 31;
  const int wave = tid >> 5;
  const int bh   = (int)blockIdx.x;   // 0 .. B*H-1
  const int b    = bh / NH;
  const int h    = bh % NH;

  __shared__ float s_tb[NT * 4];      // 200 floats: this batch's true boxes
  __shared__ float s_anch[NA * 2];    // 18 floats
  __shared__ float s_partial[8];

  // ---- TDM: broadcast the block-reused data into LDS (wave 0 issues). ----
  if (wave == 0) {
    tdm_copy_row_f32(true_boxes + (size_t)b * (NT * 4),
                     (u32)(uintptr_t)(void*)s_tb, NT * 4);
    tdm_copy_row_f32(anchors, (u32)(uintptr_t)(void*)s_anch, NA * 2);
    __builtin_amdgcn_s_wait_tensorcnt(0);   // both DMAs done
  }
  __syncthreads();

  const size_t row_base = (size_t)bh * (CELLS_PER_ROW * NCH);
  const float invG = 1.0f / (float)NW;    // grid_factor (W==H==32)
  const float invN = 1.0f / 512.0f;       // net_factor  (NET==512)

  float acc = 0.0f;

  for (int cell = wave; cell < CELLS_PER_ROW; cell += 8) {
    const int w = cell / NA;
    const int a = cell - w * NA;
    const size_t base = row_base + (size_t)cell * NCH;
    const float* pp = y_pred + base;
    const float* tt = y_true + base;

    // Lookahead prefetch for the next cell this wave will touch.
    __builtin_prefetch(pp + 8 * NCH + lane, 0, 1);
    __builtin_prefetch(tt + 8 * NCH + lane, 0, 1);

    // Lane 'lane' holds channels lane, lane+32, lane+64 (valid < 85).
    float vp0 = pp[lane];
    float vp1 = pp[lane + 32];
    float vp2 = (lane < 21) ? pp[lane + 64] : 0.0f;
    float vt0 = tt[lane];
    float vt1 = tt[lane + 32];
    float vt2 = (lane < 21) ? tt[lane + 64] : 0.0f;

    // ---- log-softmax pieces over pred classes (channels 5..84) ----
    float m = (lane >= 5) ? vp0 : -3.0e38f;
    m = fmaxf(m, vp1);
    if (lane < 21) m = fmaxf(m, vp2);
    #pragma unroll
    for (int off = 16; off; off >>= 1) m = fmaxf(m, __shfl_xor(m, off, 32));

    float s = 0.0f;
    if (lane >= 5) s += __expf(vp0 - m);
    s += __expf(vp1 - m);
    if (lane < 21) s += __expf(vp2 - m);
    #pragma unroll
    for (int off = 16; off; off >>= 1) s += __shfl_xor(s, off, 32);

    // ---- argmax over true classes (tie -> lowest channel index) ----
    float bv = (lane >= 5) ? vt0 : -3.0e38f;
    int   bc = (lane >= 5) ? lane : 0x7FFF;
    if (vt1 > bv) { bv = vt1; bc = lane + 32; }
    if (lane < 21 && vt2 > bv) { bv = vt2; bc = lane + 64; }
    #pragma unroll
    for (int off = 16; off; off >>= 1) {
      float ov = __shfl_xor(bv, off, 32);
      int   oc = __shfl_xor(bc, off, 32);
      if (ov > bv || (ov == bv && oc < bc)) { bv = ov; bc = oc; }
    }
    // Pred logit at the true class's channel (bc is wave-uniform now).
    int slot = bc >> 5;
    float cand = (slot == 0) ? vp0 : ((slot == 1) ? vp1 : vp2);
    float logit = __shfl(cand, bc & 31, 32);
    float ce = (m + __logf(s)) - logit;   // -log_softmax[true_class]

    // ---- box / confidence terms (broadcast channels 0..4) ----
    float p0 = __shfl(vp0, 0, 32), p1 = __shfl(vp0, 1, 32);
    float p2 = __shfl(vp0, 2, 32), p3 = __shfl(vp0, 3, 32);
    float p4 = __shfl(vp0, 4, 32);
    float t0 = __shfl(vt0, 0, 32), t1 = __shfl(vt0, 1, 32);
    float t2 = __shfl(vt0, 2, 32), t3 = __shfl(vt0, 3, 32);
    float om = __shfl(vt0, 4, 32);                 // object_mask == true conf

    float aw = s_anch[2 * a], ah = s_anch[2 * a + 1];
    float sig0 = 1.0f / (1.0f + __expf(-p0));
    float sig1 = 1.0f / (1.0f + __expf(-p1));
    float conf = 1.0f / (1.0f + __expf(-p4));
    float pxy_x = (float)w + sig0;
    float pxy_y = (float)h + sig1;

    float pwx = __expf(p2) * aw * invN, pwy = __expf(p3) * ah * invN;
    float pcx = pxy_x * invG, pcy = pxy_y * invG;
    float pminx = pcx - 0.5f * pwx, pmaxx = pcx + 0.5f * pwx;
    float pminy = pcy - 0.5f * pwy, pmaxy = pcy + 0.5f * pwy;
    float parea = pwx * pwy;

    // ---- best IoU over 50 true boxes: lanes cover t = lane, lane+32 ----
    float biou = 0.0f;
    #pragma unroll
    for (int k = 0; k < 2; ++k) {
      int t = lane + 32 * k;
      if (t < NT) {
        float bx = s_tb[4 * t + 0] * invG, by = s_tb[4 * t + 1] * invG;
        float bw = s_tb[4 * t + 2] * invN, bh2 = s_tb[4 * t + 3] * invN;
        float tminx = bx - 0.5f * bw,  tmaxx = bx + 0.5f * bw;
        float tminy = by - 0.5f * bh2, tmaxy = by + 0.5f * bh2;
        float iw = fmaxf(fminf(pmaxx, tmaxx) - fmaxf(pminx, tminx), 0.0f);
        float ih = fmaxf(fminf(pmaxy, tmaxy) - fmaxf(pminy, tminy), 0.0f);
        float inter = iw * ih;
        float uni = parea + bw * bh2 - inter;
        biou = fmaxf(biou, inter / uni);
      }
    }
    #pragma unroll
    for (int off = 16; off; off >>= 1) biou = fmaxf(biou, __shfl_xor(biou, off, 32));

    float cd_noobj = conf * ((biou < IGNORE_THRESH) ? 1.0f : 0.0f);
    float whs = 2.0f - (__expf(t2) * aw * invN) * (__expf(t3) * ah * invN);
    float dx = om * (pxy_x - t0) * whs;
    float dy = om * (pxy_y - t1) * whs;
    float dw = om * (p2 - t2) * whs;
    float dh = om * (p3 - t3) * whs;
    float dc = om * (conf - om) * OBJ_SCALE + (1.0f - om) * cd_noobj;

    acc += dx * dx + dy * dy + dw * dw + dh * dh + dc * dc + om * ce;
  }

  if (lane == 0) s_partial[wave] = acc;   // acc is replicated across the wave
  __syncthreads();
  if (tid == 0) {
    float t = 0.0f;
    #pragma unroll
    for (int i = 0; i < 8; ++i) t += s_partial[i];
    atomicAdd(&out[b], t);
  }
}

extern "C" void kernel_launch(void* const* d_in, const int* in_sizes, int n_in,
                              void* d_out, int out_size, void* d_ws, size_t ws_size,
                              hipStream_t stream) {
  // setup_inputs order: input_image, y_pred, y_true, true_boxes, anchors
  const float* y_pred     = (const float*)d_in[1];
  const float* y_true     = (const float*)d_in[2];
  const float* true_boxes = (const float*)d_in[3];
  const float* anchors    = (const float*)d_in[4];
  float* out = (float*)d_out;

  zero_out_kernel<<<1, 64, 0, stream>>>(out, NB);
  yolo_loss_kernel<<<NB * NH, 256, 0, stream>>>(y_pred, y_true, true_boxes,
                                                anchors, out);
}